// GCNLayer_68882685493840
// MI455X (gfx1250) — compile-verified
//
#include <hip/hip_runtime.h>
#include <hip/hip_bf16.h>
#include <math.h>

// ---------------------------------------------------------------------------
// GCN layer for MI455X (gfx1250, wave32).
// Bandwidth-bound at fp32 (~1.3-1.5 GB traffic @ 23.3 TB/s => ~60us floor);
// 45 GFLOP of matmul is negligible => stay fp32, use V_WMMA_F32_16X16X4_F32.
// Cache policy: stream E / edge_feat / node_feat (NT), keep H + out in L2.
// ---------------------------------------------------------------------------

#define NN 100000       // nodes
#define NE 640000       // edges
#define DD 128          // feature dim
#define KK2 256         // GEMM2 K (concat dim)
#define MT 32           // M rows per block (2 x 16 tiles per wave)

typedef __attribute__((ext_vector_type(2))) float v2f;
typedef __attribute__((ext_vector_type(4))) float v4f;
typedef __attribute__((ext_vector_type(8))) float v8f;

__device__ __forceinline__ float gelu_exact(float x) {
    return 0.5f * x * (1.0f + erff(x * 0.70710678118654752440f));
}

__device__ __forceinline__ double wave_sum(double v) {
#pragma unroll
    for (int off = 16; off > 0; off >>= 1) v += __shfl_down(v, off, 32);
    return v;
}

__device__ __forceinline__ v8f wmma_f32(v2f a, v2f b, v8f c) {
    return __builtin_amdgcn_wmma_f32_16x16x4_f32(false, a, false, b, (short)0,
                                                 c, false, false);
}

// ---------------------------------------------------------------------------
// zeroing (v4f grid-stride)
// ---------------------------------------------------------------------------
__global__ void zero_f32v4(v4f* __restrict__ p, long long n4) {
    long long i = (long long)blockIdx.x * blockDim.x + threadIdx.x;
    long long stride = (long long)gridDim.x * blockDim.x;
    const v4f z = {0.f, 0.f, 0.f, 0.f};
    for (; i < n4; i += stride) p[i] = z;
}

// ---------------------------------------------------------------------------
// GEMM1: H = X @ W1 + b1   (X: [NN,128], W1: [128,128])
// Block = 256 thr = 8 waves; block computes a 32x128 output stripe.
// Each wave: two 16x16 tiles sharing B fragments -> 64 wmma / wave.
// Fused: partial sum / sumsq of H -> f64 global atomics.
// ---------------------------------------------------------------------------
__global__ __launch_bounds__(256) void gemm1_kernel(
    const float* __restrict__ X, const float* __restrict__ W1,
    const float* __restrict__ b1, float* __restrict__ H,
    double* __restrict__ acc2 /* [S, SS] */) {
    __shared__ float As[MT * DD];  // 16KB A-tile (32 rows x 128)

    const int row0 = blockIdx.x * MT;
    const int tid  = threadIdx.x;

    // 32 contiguous rows: coalesced b128 streaming copy into LDS
    {
        const v4f* Xv = (const v4f*)(X + (size_t)row0 * DD);
        v4f* Av = (v4f*)As;
#pragma unroll
        for (int i = tid; i < MT * DD / 4; i += 256)
            Av[i] = __builtin_nontemporal_load(&Xv[i]);
    }
    __syncthreads();

    const int wave = tid >> 5;
    const int lane = tid & 31;
    const int half = lane >> 4;   // 0: lanes 0-15, 1: lanes 16-31
    const int lm   = lane & 15;   // M for A frag, N for B/C frags
    const int ncol = wave * 16 + lm;

    v8f acc0 = {}, acc1 = {};
#pragma unroll
    for (int k = 0; k < DD; k += 4) {
        const int ka = k + 2 * half;  // A VGPR0=K0(lo)/K2(hi), VGPR1=K1/K3
        v2f a0, a1, b;
        a0.x = As[lm * DD + ka + 0];
        a0.y = As[lm * DD + ka + 1];
        a1.x = As[(lm + 16) * DD + ka + 0];
        a1.y = As[(lm + 16) * DD + ka + 1];
        b.x = W1[(size_t)(ka + 0) * DD + ncol];
        b.y = W1[(size_t)(ka + 1) * DD + ncol];
        acc0 = wmma_f32(a0, b, acc0);
        acc1 = wmma_f32(a1, b, acc1);
    }

    const float bias = b1[ncol];
    double s = 0.0, ss = 0.0;
#pragma unroll
    for (int r = 0; r < 8; ++r) {
        const int m = r + 8 * half;  // C/D: vgpr r -> M=r (lo) / r+8 (hi)
        float v0 = acc0[r] + bias;
        float v1 = acc1[r] + bias;
        // H is gathered in GEMM2: keep regular-temporal (lives in L2)
        H[(size_t)(row0 + m) * DD + ncol] = v0;
        H[(size_t)(row0 + 16 + m) * DD + ncol] = v1;
        s += (double)v0 + (double)v1;
        ss += (double)v0 * v0 + (double)v1 * v1;
    }
    s  = wave_sum(s);
    ss = wave_sum(ss);
    if (lane == 0) {
        atomicAdd(&acc2[0], s);
        atomicAdd(&acc2[1], ss);
    }
}

// ---------------------------------------------------------------------------
// finalize LN stats: mu = S/n; rs = 1/sqrt(SS/n - mu^2 + eps)
// ---------------------------------------------------------------------------
__global__ void finalize_stats(const double* __restrict__ acc2,
                               float* __restrict__ outf, double inv_n) {
    double mu  = acc2[0] * inv_n;
    double var = acc2[1] * inv_n - mu * mu;
    outf[0] = (float)mu;
    outf[1] = (float)(1.0 / sqrt(var + 1e-5));
}

// ---------------------------------------------------------------------------
// GEMM2: E = concat(edge_feat, gelu(LN1(H[src]))) @ W2 + b2
// A-tile 32x256 in LDS (gather + LN + GELU fused into staging, b128 loads).
// Each wave: two 16x16 tiles sharing B fragments -> 128 wmma / wave.
// Fused sum/sumsq of E. E stored non-temporally (streams past L2).
// ---------------------------------------------------------------------------
__global__ __launch_bounds__(256) void gemm2_kernel(
    const float* __restrict__ EF, const float* __restrict__ H,
    const int* __restrict__ idx, const float* __restrict__ W2,
    const float* __restrict__ b2, const float* __restrict__ st1 /*mu1,rs1*/,
    float* __restrict__ E, double* __restrict__ acc2 /* [S, SS] */) {
    __shared__ float As[MT * KK2];  // 32KB A-tile (32 edges x 256)
    __shared__ int srcs[MT];

    const int e0  = blockIdx.x * MT;
    const int tid = threadIdx.x;
    const float mu1 = st1[0];
    const float rs1 = st1[1];

    if (tid < MT) srcs[tid] = idx[(size_t)(e0 + tid) * 2 + 0];
    __syncthreads();

    // stage A-tile as v4f: 32*64 = 2048 vectors, 8 per thread
#pragma unroll
    for (int i4 = tid; i4 < MT * (KK2 / 4); i4 += 256) {
        const int r  = i4 >> 6;   // edge row (64 v4f per row)
        const int c4 = i4 & 63;   // v4f column
        v4f v;
        if (c4 < DD / 4) {
            // edge features: read-once -> non-temporal
            v = __builtin_nontemporal_load(
                (const v4f*)(EF + (size_t)(e0 + r) * DD) + c4);
        } else {
            // gathered h: random access, L2-resident -> regular temporal
            const int s = srcs[r];
            v4f h4 = *((const v4f*)(H + (size_t)s * DD) + (c4 - DD / 4));
            v.x = gelu_exact((h4.x - mu1) * rs1);
            v.y = gelu_exact((h4.y - mu1) * rs1);
            v.z = gelu_exact((h4.z - mu1) * rs1);
            v.w = gelu_exact((h4.w - mu1) * rs1);
        }
        ((v4f*)As)[i4] = v;
    }
    __syncthreads();

    const int wave = tid >> 5;
    const int lane = tid & 31;
    const int half = lane >> 4;
    const int lm   = lane & 15;
    const int ncol = wave * 16 + lm;

    v8f acc0 = {}, acc1 = {};
#pragma unroll 4
    for (int k = 0; k < KK2; k += 4) {
        const int ka = k + 2 * half;
        v2f a0, a1, b;
        a0.x = As[lm * KK2 + ka + 0];
        a0.y = As[lm * KK2 + ka + 1];
        a1.x = As[(lm + 16) * KK2 + ka + 0];
        a1.y = As[(lm + 16) * KK2 + ka + 1];
        b.x = W2[(size_t)(ka + 0) * DD + ncol];
        b.y = W2[(size_t)(ka + 1) * DD + ncol];
        acc0 = wmma_f32(a0, b, acc0);
        acc1 = wmma_f32(a1, b, acc1);
    }

    const float bias = b2[ncol];
    double s = 0.0, ss = 0.0;
#pragma unroll
    for (int r = 0; r < 8; ++r) {
        const int m = r + 8 * half;
        float v0 = acc0[r] + bias;
        float v1 = acc1[r] + bias;
        // E streams (328MB write-once): non-temporal, keep L2 for H/out
        __builtin_nontemporal_store(v0, &E[(size_t)(e0 + m) * DD + ncol]);
        __builtin_nontemporal_store(v1, &E[(size_t)(e0 + 16 + m) * DD + ncol]);
        s += (double)v0 + (double)v1;
        ss += (double)v0 * v0 + (double)v1 * v1;
    }
    s  = wave_sum(s);
    ss = wave_sum(ss);
    if (lane == 0) {
        atomicAdd(&acc2[0], s);
        atomicAdd(&acc2[1], ss);
    }
}

// ---------------------------------------------------------------------------
// scatter: out[dst] += gelu(LN2(E_row)); counts[dst] += 1
// One wave per edge row; lane handles one v4f. Atomics hit L2 (51MB out).
// ---------------------------------------------------------------------------
__global__ __launch_bounds__(256) void scatter_kernel(
    const float* __restrict__ E, const int* __restrict__ idx,
    const float* __restrict__ st2 /*mu2,rs2*/, float* __restrict__ out,
    float* __restrict__ counts) {
    const int gid  = blockIdx.x * 256 + threadIdx.x;
    const int edge = gid >> 5;
    const int lane = gid & 31;
    if (edge >= NE) return;

    const float mu2 = st2[0];
    const float rs2 = st2[1];
    const int dst = idx[(size_t)edge * 2 + 1];

    // last use of E: non-temporal
    const v4f v = __builtin_nontemporal_load(
        ((const v4f*)(E + (size_t)edge * DD)) + lane);
    float gx = gelu_exact((v.x - mu2) * rs2);
    float gy = gelu_exact((v.y - mu2) * rs2);
    float gz = gelu_exact((v.z - mu2) * rs2);
    float gw = gelu_exact((v.w - mu2) * rs2);

    float* o = out + (size_t)dst * DD + lane * 4;
    __hip_atomic_fetch_add(&o[0], gx, __ATOMIC_RELAXED, __HIP_MEMORY_SCOPE_AGENT);
    __hip_atomic_fetch_add(&o[1], gy, __ATOMIC_RELAXED, __HIP_MEMORY_SCOPE_AGENT);
    __hip_atomic_fetch_add(&o[2], gz, __ATOMIC_RELAXED, __HIP_MEMORY_SCOPE_AGENT);
    __hip_atomic_fetch_add(&o[3], gw, __ATOMIC_RELAXED, __HIP_MEMORY_SCOPE_AGENT);
    if (lane == 0)
        __hip_atomic_fetch_add(&counts[dst], 1.0f, __ATOMIC_RELAXED,
                               __HIP_MEMORY_SCOPE_AGENT);
}

// ---------------------------------------------------------------------------
// out = counts > 0 ? out / counts : 0   (v4f per thread)
// ---------------------------------------------------------------------------
__global__ __launch_bounds__(256) void divide_kernel(
    v4f* __restrict__ out, const float* __restrict__ counts) {
    const long long gid = (long long)blockIdx.x * 256 + threadIdx.x;
    if (gid >= (long long)NN * (DD / 4)) return;
    const int node = (int)(gid >> 5);  // 32 v4f per row
    const float c = counts[node];
    v4f v = out[gid];
    if (c > 0.0f) {
        const float inv = 1.0f / c;
        v *= inv;
    } else {
        v = (v4f){0.f, 0.f, 0.f, 0.f};
    }
    out[gid] = v;
}

// ---------------------------------------------------------------------------
// launch
// ---------------------------------------------------------------------------
extern "C" void kernel_launch(void* const* d_in, const int* in_sizes, int n_in,
                              void* d_out, int out_size, void* d_ws,
                              size_t ws_size, hipStream_t stream) {
    const float* edge_features = (const float*)d_in[0];  // [NE, 128]
    const float* node_features = (const float*)d_in[1];  // [NN, 128]
    const int*   edge_indices  = (const int*)d_in[2];    // [NE, 2]
    const float* W1            = (const float*)d_in[3];  // [128, 128]
    const float* b1            = (const float*)d_in[4];  // [128]
    const float* W2            = (const float*)d_in[5];  // [256, 128]
    const float* b2            = (const float*)d_in[6];  // [128]
    float* out = (float*)d_out;                          // [NN, 128]

    // workspace layout (floats):
    //   [0, NN*DD)                H
    //   [.., +NE*DD)              E
    //   [.., +NN)                 counts
    //   4 doubles (8-aligned):    S1, SS1, S2, SS2
    //   4 floats:                 mu1, rs1, mu2, rs2
    float* ws = (float*)d_ws;
    const size_t E_off = (size_t)NN * DD;                  // 12,800,000
    const size_t C_off = E_off + (size_t)NE * DD;          // 94,720,000
    const size_t D_off = C_off + (size_t)NN;               // 94,820,000
    const size_t F_off = D_off + 8;

    float*  H      = ws;
    float*  E      = ws + E_off;
    float*  counts = ws + C_off;
    double* acc    = (double*)(ws + D_off);  // acc[0..3]
    float*  stf    = ws + F_off;             // stf[0..3]

    // 1) zero output, counts, and the 4 double accumulators
    zero_f32v4<<<2048, 256, 0, stream>>>((v4f*)out, (long long)NN * DD / 4);
    zero_f32v4<<<128, 256, 0, stream>>>((v4f*)counts,
                                        (long long)(NN + 8) / 4);

    // 2) GEMM1 + h stats (100000/32 = 3125 stripes)
    gemm1_kernel<<<NN / MT, 256, 0, stream>>>(node_features, W1, b1, H, acc);

    // 3) finalize mu1/rs1
    finalize_stats<<<1, 1, 0, stream>>>(acc, stf, 1.0 / ((double)NN * DD));

    // 4) GEMM2 (gather + LN1 + GELU fused) + e stats (640000/32 = 20000)
    gemm2_kernel<<<NE / MT, 256, 0, stream>>>(edge_features, H, edge_indices,
                                              W2, b2, stf, E, acc + 2);

    // 5) finalize mu2/rs2
    finalize_stats<<<1, 1, 0, stream>>>(acc + 2, stf + 2,
                                        1.0 / ((double)NE * DD));

    // 6) LN2 + GELU + scatter-add (one wave per edge row)
    scatter_kernel<<<(NE * 32) / 256, 256, 0, stream>>>(E, edge_indices,
                                                        stf + 2, out, counts);

    // 7) segment mean
    divide_kernel<<<((long long)NN * (DD / 4) + 255) / 256, 256, 0, stream>>>(
        (v4f*)out, counts);
}